// DCNv3_pytorch_49538152792437
// MI455X (gfx1250) — compile-verified
//
#include <hip/hip_runtime.h>
#include <math.h>

// ---------------- problem constants ----------------
#define NB_IMG 8
#define HH     64
#define WW     64
#define CDIM   128
#define GRP    8
#define GC     16
#define PP     9
#define HIN    66
#define WIN    66
#define NPIX   (NB_IMG * HH * WW)        // 32768

// ---------------- workspace layout (floats) ----------------
#define XPAD_OFF 0L
#define XPAD_N   ((long)NB_IMG * HIN * WIN * CDIM)     // 4,460,544
#define X1_OFF   (XPAD_OFF + XPAD_N)
#define X1_N     ((long)NPIX * CDIM)                   // 4,194,304
#define OFF_OFF  (X1_OFF + X1_N)
#define OFF_N    ((long)NPIX * 144)                    // 4,718,592
#define MSK_OFF  (OFF_OFF + OFF_N)
#define MSK_N    ((long)NPIX * 80)                     // 2,621,440
#define SMP_OFF  (MSK_OFF + MSK_N)
#define SMP_N    ((long)NPIX * CDIM)                   // 4,194,304
#define WMP_OFF  (SMP_OFF + SMP_N)
#define WMP_N    (128L * 80)
#define BMP_OFF  (WMP_OFF + WMP_N)

typedef __attribute__((ext_vector_type(2))) float v2f;
typedef __attribute__((ext_vector_type(8))) float v8f;

// =====================================================================
// Generic FP32 WMMA GEMM:  C[M x nvalid] = A[M x 128] * B[128 x NB] + bias
// 256 threads = 8 waves; wave w computes the 16x16 tile at colTile
// (blockIdx.y*8 + w).  A tile and a 128-wide B panel staged in LDS.
// PADSTORE==1 scatters rows into the zero-padded (8,66,66,128) image.
// =====================================================================
template <int PADSTORE>
__global__ __launch_bounds__(256) void gemm_wmma_f32(
    const float* __restrict__ A, const float* __restrict__ B,
    const float* __restrict__ bias, float* __restrict__ C,
    int colTiles, int nvalid, int NBcols, int ldc) {
  __shared__ float As[16][132];   // pad 132: A-frag gather conflict-free
  __shared__ float Bs[128][136];  // pad 136: B-frag gather conflict-free

  const int tid   = threadIdx.x;
  const int wave  = tid >> 5;
  const int lane  = tid & 31;
  const int l16   = lane & 15;
  const int hi    = lane >> 4;    // 0 or 1 (half-wave)
  const int khalf = hi << 1;      // ISA 32-bit A/B layout: half-wave K offset
  const int rowBase   = blockIdx.x << 4;
  const int colTile   = (blockIdx.y << 3) + wave;
  const int colBase   = colTile << 4;
  const int panelBase = blockIdx.y << 7;   // first global col in this Bs panel

  // ---- stage A tile: 16 x 128 floats, 8 floats per thread ----
  {
    const int e = tid << 3;
    const int r = e >> 7;
    const int c = e & 127;
    const float4* src = (const float4*)(A + (size_t)(rowBase + r) * CDIM + c);
    *(float4*)&As[r][c]     = src[0];
    *(float4*)&As[r][c + 4] = src[1];
  }
  // ---- stage B panel: 128 x 128 floats (guarded, zero-fill OOB cols) ----
#pragma unroll
  for (int it = 0; it < 16; ++it) {
    const int idx = (it << 8) + tid;       // float4 index 0..4095
    const int e   = idx << 2;
    const int r   = e >> 7;
    const int c   = e & 127;
    const int gcol = panelBase + c;
    float4 v = make_float4(0.f, 0.f, 0.f, 0.f);
    if (gcol + 4 <= NBcols) v = *(const float4*)(B + (size_t)r * NBcols + gcol);
    *(float4*)&Bs[r][c] = v;
  }
  __syncthreads();

  // ---- accumulator init with bias (D-tile layout: N = l16 for all 8 regs) ----
  v8f acc;
  float bv = 0.f;
  if (colTile < colTiles && (colBase + l16) < nvalid) bv = bias[colBase + l16];
#pragma unroll
  for (int r = 0; r < 8; ++r) acc[r] = bv;

  if (colTile < colTiles) {   // wave-uniform: EXEC stays all-ones for WMMA
    const int lcol = (colBase - panelBase) + l16;
#pragma unroll
    for (int k0 = 0; k0 < 128; k0 += 4) {
      v2f a, b;
      // 32-bit A 16x4: v0 = K (lanes0-15) / K+2 (lanes16-31), v1 = +1
      a.x = As[l16][k0 + khalf];
      a.y = As[l16][k0 + khalf + 1];
      // 32-bit B 4x16: row-of-B striped across lanes, same half-wave split
      b.x = Bs[k0 + khalf][lcol];
      b.y = Bs[k0 + khalf + 1][lcol];
      acc = __builtin_amdgcn_wmma_f32_16x16x4_f32(
          false, a, false, b, (short)0, acc, false, false);
    }
    // ---- store: reg r -> M = r + 8*hi, N = l16 ----
#pragma unroll
    for (int r = 0; r < 8; ++r) {
      const int row = rowBase + r + (hi << 3);
      const int col = colBase + l16;
      if (col < nvalid) {
        size_t addr;
        if (PADSTORE) {
          const int n = row >> 12, rem = row & 4095;
          const int h = rem >> 6, w = rem & 63;
          addr = ((size_t)((n * HIN + h + 1) * WIN + (w + 1))) * CDIM + col;
        } else {
          addr = (size_t)row * ldc + col;
        }
        C[addr] = acc[r];
      }
    }
  }
}

// =====================================================================
// depthwise 3x3 conv + LayerNorm(C) + exact GELU ; one pixel per block
// =====================================================================
__global__ __launch_bounds__(128) void dwln_gelu(
    const float* __restrict__ x, const float* __restrict__ dww,
    const float* __restrict__ dwb, const float* __restrict__ lng,
    const float* __restrict__ lnb, float* __restrict__ x1) {
  const int pix = blockIdx.x;
  const int c   = threadIdx.x;
  const int n = pix >> 12, rem = pix & 4095;
  const int h = rem >> 6, w = rem & 63;

  float v = dwb[c];
#pragma unroll
  for (int kh = 0; kh < 3; ++kh) {
    const int hh = h + kh - 1;
    if ((unsigned)hh >= (unsigned)HH) continue;
#pragma unroll
    for (int kw = 0; kw < 3; ++kw) {
      const int ww2 = w + kw - 1;
      if ((unsigned)ww2 >= (unsigned)WW) continue;
      v = fmaf(x[(((size_t)n * HH + hh) * WW + ww2) * CDIM + c],
               dww[(kh * 3 + kw) * CDIM + c], v);
    }
  }
  // block reduction (4 wave32's) for mean / mean-of-squares
  __shared__ float red[8];
  float s = v, s2 = v * v;
#pragma unroll
  for (int off = 16; off; off >>= 1) {
    s  += __shfl_down(s, off, 32);
    s2 += __shfl_down(s2, off, 32);
  }
  const int wv = threadIdx.x >> 5, ln = threadIdx.x & 31;
  if (ln == 0) { red[wv] = s; red[wv + 4] = s2; }
  __syncthreads();
  const float ts  = red[0] + red[1] + red[2] + red[3];
  const float ts2 = red[4] + red[5] + red[6] + red[7];
  const float mu  = ts * (1.f / 128.f);
  const float var = ts2 * (1.f / 128.f) - mu * mu;
  float xn = (v - mu) * rsqrtf(var + 1e-6f) * lng[c] + lnb[c];
  // exact GELU (approximate=False): 0.5 x (1 + erf(x/sqrt(2)))
  xn = 0.5f * xn * (1.f + erff(xn * 0.70710678118654752440f));
  x1[(size_t)pix * CDIM + c] = xn;
}

// =====================================================================
// per-(pixel, group) softmax over the 9 sampling points (in place)
// =====================================================================
__global__ __launch_bounds__(256) void softmax9(float* __restrict__ msk) {
  const int i = blockIdx.x * 256 + threadIdx.x;
  if (i >= NPIX * GRP) return;
  const int row = i >> 3, g = i & 7;
  float* p = msk + (size_t)row * 80 + g * 9;
  float m[9], mx = -3.4e38f;
#pragma unroll
  for (int k = 0; k < 9; ++k) { m[k] = p[k]; mx = fmaxf(mx, m[k]); }
  float sum = 0.f;
#pragma unroll
  for (int k = 0; k < 9; ++k) { m[k] = expf(m[k] - mx); sum += m[k]; }
  const float inv = 1.f / sum;
#pragma unroll
  for (int k = 0; k < 9; ++k) p[k] = m[k] * inv;
}

// =====================================================================
// deformable bilinear sampling; one thread per (pixel, group)
// =====================================================================
__device__ __forceinline__ void tap16(const float* __restrict__ base,
                                      float wgt, float acc[16]) {
#pragma unroll
  for (int c4 = 0; c4 < 4; ++c4) {
    const float4 v = *(const float4*)(base + (c4 << 2));
    acc[c4 * 4 + 0] = fmaf(wgt, v.x, acc[c4 * 4 + 0]);
    acc[c4 * 4 + 1] = fmaf(wgt, v.y, acc[c4 * 4 + 1]);
    acc[c4 * 4 + 2] = fmaf(wgt, v.z, acc[c4 * 4 + 2]);
    acc[c4 * 4 + 3] = fmaf(wgt, v.w, acc[c4 * 4 + 3]);
  }
}

__global__ __launch_bounds__(256) void dcn_sample(
    const float* __restrict__ xpad, const float* __restrict__ offs,
    const float* __restrict__ msk, float* __restrict__ smp) {
  const int i = blockIdx.x * 256 + threadIdx.x;
  if (i >= NPIX * GRP) return;
  const int g = i & 7, row = i >> 3;
  const int n = row >> 12, rem = row & 4095;
  const int h = rem >> 6, w = rem & 63;

  const float* op = offs + (size_t)row * 144 + g * 18;  // (x,y) pairs
  const float* mp = msk + (size_t)row * 80 + g * 9;
  const float* img = xpad + (size_t)n * HIN * WIN * CDIM + g * GC;

  float acc[16];
#pragma unroll
  for (int c = 0; c < 16; ++c) acc[c] = 0.f;

#pragma unroll
  for (int p = 0; p < PP; ++p) {
    // reference: pts x-major flatten, grid math reduces to padded coords
    const float dx = (float)(p / 3 - 1);
    const float dy = (float)(p % 3 - 1);
    const float px = (float)(w + 1) + dx + op[2 * p];
    const float py = (float)(h + 1) + dy + op[2 * p + 1];
    const float fx = floorf(px), fy = floorf(py);
    const float wx = px - fx, wy = py - fy;
    const int x0 = (int)fx, y0 = (int)fy;
    const float mv = mp[p];
    const float w00 = (1.f - wx) * (1.f - wy) * mv;
    const float w10 = wx * (1.f - wy) * mv;
    const float w01 = (1.f - wx) * wy * mv;
    const float w11 = wx * wy * mv;
    const bool vx0 = (unsigned)x0 < (unsigned)WIN;
    const bool vx1 = (unsigned)(x0 + 1) < (unsigned)WIN;
    const bool vy0 = (unsigned)y0 < (unsigned)HIN;
    const bool vy1 = (unsigned)(y0 + 1) < (unsigned)HIN;
    if (vy0 && vx0) tap16(img + ((size_t)y0 * WIN + x0) * CDIM, w00, acc);
    if (vy0 && vx1) tap16(img + ((size_t)y0 * WIN + x0 + 1) * CDIM, w10, acc);
    if (vy1 && vx0) tap16(img + ((size_t)(y0 + 1) * WIN + x0) * CDIM, w01, acc);
    if (vy1 && vx1) tap16(img + ((size_t)(y0 + 1) * WIN + x0 + 1) * CDIM, w11, acc);
  }
  float4* out = (float4*)(smp + (size_t)row * CDIM + g * GC);
#pragma unroll
  for (int c4 = 0; c4 < 4; ++c4) {
    out[c4] = make_float4(acc[c4 * 4 + 0], acc[c4 * 4 + 1],
                          acc[c4 * 4 + 2], acc[c4 * 4 + 3]);
  }
}

// =====================================================================
// helpers: zero the padded image, pad mask weights 72 -> 80 cols
// =====================================================================
__global__ void zero_ws(float* __restrict__ p, long nfl) {
  long i = (long)blockIdx.x * blockDim.x + threadIdx.x;
  const long stride = (long)gridDim.x * blockDim.x;
  for (; i < nfl; i += stride) p[i] = 0.f;
}

__global__ void pack_mask_w(const float* __restrict__ wm,
                            const float* __restrict__ bm,
                            float* __restrict__ wmp, float* __restrict__ bmp) {
  const int i = blockIdx.x * 256 + threadIdx.x;
  if (i < 128 * 80) {
    const int r = i / 80, c = i % 80;
    wmp[i] = (c < 72) ? wm[r * 72 + c] : 0.f;
  }
  if (i < 80) bmp[i] = (i < 72) ? bm[i] : 0.f;
}

// =====================================================================
extern "C" void kernel_launch(void* const* d_in, const int* in_sizes, int n_in,
                              void* d_out, int out_size, void* d_ws,
                              size_t ws_size, hipStream_t stream) {
  const float* x     = (const float*)d_in[0];
  const float* w_in  = (const float*)d_in[1];
  const float* b_in  = (const float*)d_in[2];
  const float* dw_w  = (const float*)d_in[3];
  const float* dw_b  = (const float*)d_in[4];
  const float* ln_g  = (const float*)d_in[5];
  const float* ln_b  = (const float*)d_in[6];
  const float* w_off = (const float*)d_in[7];
  const float* b_off = (const float*)d_in[8];
  const float* w_msk = (const float*)d_in[9];
  const float* b_msk = (const float*)d_in[10];
  const float* w_out = (const float*)d_in[11];
  const float* b_out = (const float*)d_in[12];
  float* out = (float*)d_out;

  float* ws   = (float*)d_ws;
  float* xpad = ws + XPAD_OFF;
  float* x1   = ws + X1_OFF;
  float* offs = ws + OFF_OFF;
  float* msk  = ws + MSK_OFF;
  float* smp  = ws + SMP_OFF;
  float* wmp  = ws + WMP_OFF;
  float* bmp  = ws + BMP_OFF;

  const int rowTiles = NPIX / 16;  // 2048

  // 0) zero padded image borders + pack padded mask weights
  zero_ws<<<2048, 256, 0, stream>>>(xpad, XPAD_N);
  pack_mask_w<<<(128 * 80 + 255) / 256, 256, 0, stream>>>(w_msk, b_msk, wmp, bmp);

  // 1) x_proj = x @ w_in + b_in, scattered into padded (8,66,66,128)
  gemm_wmma_f32<1><<<dim3(rowTiles, 1), 256, 0, stream>>>(
      x, w_in, b_in, xpad, /*colTiles=*/8, /*nvalid=*/128, /*NB=*/128, 128);

  // 2) x1 = GELU(LN(dwconv(x)))
  dwln_gelu<<<NPIX, 128, 0, stream>>>(x, dw_w, dw_b, ln_g, ln_b, x1);

  // 3) offset = x1 @ w_off + b_off   (N = 144 = 9 col tiles -> gridDim.y = 2)
  gemm_wmma_f32<0><<<dim3(rowTiles, 2), 256, 0, stream>>>(
      x1, w_off, b_off, offs, /*colTiles=*/9, /*nvalid=*/144, /*NB=*/144, 144);

  // 4) mask logits = x1 @ w_msk_pad + b_msk_pad  (72 valid, stride 80)
  gemm_wmma_f32<0><<<dim3(rowTiles, 1), 256, 0, stream>>>(
      x1, wmp, bmp, msk, /*colTiles=*/5, /*nvalid=*/72, /*NB=*/80, 80);

  // 5) softmax over the 9 points per (pixel, group)
  softmax9<<<(NPIX * GRP + 255) / 256, 256, 0, stream>>>(msk);

  // 6) deformable bilinear sampling -> (pixel, g*16+c) layout
  dcn_sample<<<(NPIX * GRP + 255) / 256, 256, 0, stream>>>(xpad, offs, msk, smp);

  // 7) out = sampled @ w_out + b_out
  gemm_wmma_f32<0><<<dim3(rowTiles, 1), 256, 0, stream>>>(
      smp, w_out, b_out, out, /*colTiles=*/8, /*nvalid=*/128, /*NB=*/128, 128);
}